// FFM_Layer_52467320488379
// MI455X (gfx1250) — compile-verified
//
#include <hip/hip_runtime.h>

typedef float v2f __attribute__((ext_vector_type(2)));
typedef float v4f __attribute__((ext_vector_type(4)));
typedef float v8f __attribute__((ext_vector_type(8)));

#define N_DENSE   13
#define N_SPARSE  26
#define FEAT_NUM  10000
#define COLS      312      // FIELD_NUM * K = 39*8 (true row stride of v)
#define COLS_PAD  320      // NTILES*16, zero-padded in LDS
#define NTILES    20
#define ROWS_PER_WAVE  16
#define WAVES_PER_BLOCK 2
#define BLOCK_THREADS  64
#define ROWS_PER_BLOCK 32

struct __align__(16) WaveShm {
    float G[ROWS_PER_WAVE][COLS_PAD];  // gather-sum, cols 312..319 kept zero
    float Dls[ROWS_PER_WAVE][16];      // dense rows, cols 13..15 kept zero
    float F[ROWS_PER_WAVE];            // first-order per row
    int   gix[ROWS_PER_WAVE][N_SPARSE];
};

__global__ __launch_bounds__(BLOCK_THREADS)
void ffm_layer_kernel(const float* __restrict__ dense,
                      const int*   __restrict__ sparse,
                      const float* __restrict__ w0p,
                      const float* __restrict__ w,
                      const float* __restrict__ v,
                      float*       __restrict__ out)
{
    __shared__ WaveShm shm[WAVES_PER_BLOCK];
    const int lane = threadIdx.x & 31;
    const int wave = threadIdx.x >> 5;
    WaveShm& S = shm[wave];
    const int r0 = blockIdx.x * ROWS_PER_BLOCK + wave * ROWS_PER_WAVE;

    // ---- Phase 0: stage sparse ids (+field offsets), dense rows, zero pads ----
    for (int t = lane; t < ROWS_PER_WAVE * N_SPARSE; t += 32) {
        int row = t / N_SPARSE, s = t - row * N_SPARSE;
        int id = sparse[(r0 + row) * N_SPARSE + s];
        S.gix[row][s] = N_DENSE + FEAT_NUM * s + id;
    }
    for (int t = lane; t < ROWS_PER_WAVE * N_DENSE; t += 32) {
        int row = t / N_DENSE, d = t - row * N_DENSE;
        S.Dls[row][d] = dense[(r0 + row) * N_DENSE + d];
    }
    for (int t = lane; t < ROWS_PER_WAVE * 3; t += 32)        // Dls pad = 0
        S.Dls[t / 3][N_DENSE + (t % 3)] = 0.f;
    for (int t = lane; t < ROWS_PER_WAVE * 8; t += 32)        // G col pad = 0
        S.G[t >> 3][COLS + (t & 7)] = 0.f;
    __syncthreads();

    // ---- Phase 1: gather-accumulate Sum_s v[gidx] into G (b128 gathers) ----
    const int c2 = (lane + 64 < 78) ? (lane + 64) : 77;  // clamp: branch-free loop
    for (int row = 0; row < ROWS_PER_WAVE; ++row) {
        v4f a0 = {0.f,0.f,0.f,0.f}, a1 = {0.f,0.f,0.f,0.f}, a2 = {0.f,0.f,0.f,0.f};
        for (int s = 0; s < N_SPARSE; ++s) {
            const float* rowp = v + (long long)S.gix[row][s] * COLS;
            if (s + 1 < N_SPARSE) {
                // spread lanes across the next 1248B row -> global_prefetch_b8
                const float* np = v + (long long)S.gix[row][s + 1] * COLS + lane * 10;
                __builtin_prefetch(np, 0, 0);
            }
            const v4f* p = (const v4f*)rowp;          // 78 float4 chunks, 16B aligned
            a0 += p[lane];
            a1 += p[lane + 32];
            a2 += p[c2];                              // dup loads coalesce; junk unused
        }
        v4f* g = (v4f*)&S.G[row][0];                  // row stride 1280B, 16B aligned
        g[lane] = a0;
        g[lane + 32] = a1;
        if (lane < 14) g[lane + 64] = a2;             // chunks 78,79 stay zero (pad)
    }
    // first order: lane m handles row m
    if (lane < ROWS_PER_WAVE) {
        int row = lane;
        float fo = w0p[0];
        #pragma unroll
        for (int d = 0; d < N_DENSE; ++d) fo += S.Dls[row][d] * w[d];
        for (int s = 0; s < N_SPARSE; ++s) fo += w[S.gix[row][s]];
        S.F[row] = fo;
    }
    __syncthreads();

    // ---- Phase 2: dense GEMM via V_WMMA_F32_16X16X4_F32, fused reductions ----
    const int half = lane >> 4;   // row-half select per C layout
    const int lp   = lane & 15;   // column within tile / A-matrix row

    // A operands (16x4 f32 layout: VGPR0=K even, VGPR1=K odd; halves hold K, K+2).
    // Zero padding of Dls cols 13..15 makes ALL d>=13 K-slots multiply by 0,
    // so B never needs masking anywhere below.
    v2f A[4];
    #pragma unroll
    for (int t = 0; t < 4; ++t) {
        int d0 = 4 * t + 2 * half;                    // <= 15: padded Dls is safe
        A[t].x = S.Dls[lp][d0];
        A[t].y = S.Dls[lp][d0 + 1];
    }

    float svec[8] = {0.f,0.f,0.f,0.f,0.f,0.f,0.f,0.f};
    float ssq [8] = {0.f,0.f,0.f,0.f,0.f,0.f,0.f,0.f};

    // Hot loop: tiles 0..18 are fully valid (cols < 304 < 312). No masks at all.
    for (int n = 0; n < NTILES - 1; ++n) {
        int col = 16 * n + lp;
        v8f c;
        #pragma unroll
        for (int j = 0; j < 8; ++j)                   // unconditional ds loads
            c[j] = S.G[8 * half + j][col];
        #pragma unroll
        for (int t = 0; t < 4; ++t) {
            int d0 = 4 * t + 2 * half;
            v2f B;                                    // unconditional global loads;
            B.x = v[d0 * COLS + col];                 // rows 13..15 data is killed
            B.y = v[(d0 + 1) * COLS + col];           // by the zero A values
            c = __builtin_amdgcn_wmma_f32_16x16x4_f32(
                    false, A[t], false, B, (short)0, c, false, false);
        }
        #pragma unroll
        for (int j = 0; j < 8; ++j) { float x = c[j]; svec[j] += x; ssq[j] += x * x; }
    }

    // Peeled tile 19: cols 304..319; guard only the reduction (pure v_cndmask).
    {
        int col = 16 * (NTILES - 1) + lp;
        bool cv = (col < COLS);
        v8f c;
        #pragma unroll
        for (int j = 0; j < 8; ++j)
            c[j] = S.G[8 * half + j][col];            // pad cols read 0
        #pragma unroll
        for (int t = 0; t < 4; ++t) {
            int d0 = 4 * t + 2 * half;
            v2f B;
            B.x = v[d0 * COLS + col];                 // still in-bounds of v
            B.y = v[(d0 + 1) * COLS + col];
            c = __builtin_amdgcn_wmma_f32_16x16x4_f32(
                    false, A[t], false, B, (short)0, c, false, false);
        }
        #pragma unroll
        for (int j = 0; j < 8; ++j) {
            float x = cv ? c[j] : 0.f;                // register select, no load
            svec[j] += x; ssq[j] += x * x;
        }
    }

    // wave32 butterfly reductions (masks stay inside each 16-lane half)
    #pragma unroll
    for (int j = 0; j < 8; ++j) {
        float sv = svec[j];
        sv += __shfl_xor(sv, 8, 32);          // combine lanes k and k+8 -> s_vec[k]
        float s2 = sv * sv;                   // s_vec[k]^2
        s2 += __shfl_xor(s2, 1, 32);
        s2 += __shfl_xor(s2, 2, 32);
        s2 += __shfl_xor(s2, 4, 32);          // Sum_k s_vec[k]^2, per row 8*half+j
        float q = ssq[j];
        q += __shfl_xor(q, 1, 32);
        q += __shfl_xor(q, 2, 32);
        q += __shfl_xor(q, 4, 32);
        q += __shfl_xor(q, 8, 32);            // Sum_{f,k} field^2, per row
        svec[j] = 0.5f * (s2 - q);            // second_order for row 8*half+j
    }

    if (lp == 0) {                             // lanes 0 and 16 write 8 rows each
        #pragma unroll
        for (int j = 0; j < 8; ++j) {
            int m = 8 * half + j;
            out[r0 + m] = S.F[m] + svec[j];
        }
    }
}

extern "C" void kernel_launch(void* const* d_in, const int* in_sizes, int n_in,
                              void* d_out, int out_size, void* d_ws, size_t ws_size,
                              hipStream_t stream) {
    const float* dense  = (const float*)d_in[0];
    const int*   sparse = (const int*)  d_in[1];
    const float* w0     = (const float*)d_in[2];
    const float* w      = (const float*)d_in[3];
    const float* v      = (const float*)d_in[4];
    float* out = (float*)d_out;

    const int batch = out_size;                 // 4096
    dim3 grid((batch + ROWS_PER_BLOCK - 1) / ROWS_PER_BLOCK);
    ffm_layer_kernel<<<grid, BLOCK_THREADS, 0, stream>>>(dense, sparse, w0, w, v, out);
}